// FNOGNO_71622874628467
// MI455X (gfx1250) — compile-verified
//
#include <hip/hip_runtime.h>
#include <math.h>

#define DD 48
#define SS 54
#define S3 157464          // 54^3
#define S3P 157696         // 256 * 616
#define CP 96
#define CH 86
#define NIN 8192
#define KNB 24
#define NOUT 110592        // 48^3
#define NT_COLS 256
#define NBLK_GEMM 616      // S3P / NT_COLS

typedef float v2f __attribute__((ext_vector_type(2)));
typedef float v8f __attribute__((ext_vector_type(8)));

__device__ __forceinline__ v8f wmma4(v2f a, v2f b, v8f c) {
  return __builtin_amdgcn_wmma_f32_16x16x4_f32(false, a, false, b, (short)0, c, false, false);
}

__device__ __forceinline__ float gelu_f(float x) {
  return 0.5f * x * (1.0f + erff(x * 0.70710678118654752440f));
}

// ---------------- twiddles: tw[pos][k] = (cos,sin)(2*pi*pos*freq(k)/54), freq = {0..5, 48..53}
__global__ void twiddle_init_k(float* tw) {
  int i = blockIdx.x * 256 + threadIdx.x;
  if (i >= SS * 12) return;
  int pos = i / 12, k = i % 12;
  int freq = (k < 6) ? k : (SS - 12 + k);
  float th = 6.28318530717958647692f * (float)(pos * freq) / (float)SS;
  tw[2 * i] = cosf(th);
  tw[2 * i + 1] = sinf(th);
}

// ---------------- positional embedding
__device__ __forceinline__ void pos_embed16(float v, float* dst) {
#pragma unroll
  for (int j = 0; j < 8; ++j) {
    float f = expf(-1.15129254649702284201f * (float)j); // 10000^(-j/8)
    float a = v * f;
    dst[j] = cosf(a);
    dst[8 + j] = sinf(a);
  }
}

__global__ void embed_out_k(float* e) {
  int p = blockIdx.x * 256 + threadIdx.x;
  if (p >= NOUT) return;
  int x = p / (DD * DD), y = (p / DD) % DD, z = p % DD;
  const float inv = 1.0f / 47.0f;
  float* d = e + (size_t)p * 48;
  pos_embed16(x * inv, d);
  pos_embed16(y * inv, d + 16);
  pos_embed16(z * inv, d + 32);
}

__global__ void embed_in_k(const float* __restrict__ xin, float* e) {
  int p = blockIdx.x * 256 + threadIdx.x;
  if (p >= NIN) return;
  float* d = e + (size_t)p * 48;
  pos_embed16(xin[p * 3 + 0], d);
  pos_embed16(xin[p * 3 + 1], d + 16);
  pos_embed16(xin[p * 3 + 2], d + 32);
}

// ---------------- transpose [N][K] -> [K][NP] with zero pad
__global__ void transpose_pad_k(const float* __restrict__ src, float* __restrict__ dst,
                                int N, int K, int NP) {
  int i = blockIdx.x * 256 + threadIdx.x;
  if (i >= K * NP) return;
  int k = i / NP, n = i % NP;
  dst[i] = (n < N) ? src[(size_t)n * K + k] : 0.0f;
}

// ---------------- lift: 5 -> 86 channels over D^3 into padded S^3 buffer
__global__ void lift_k(const float* __restrict__ df, const float* __restrict__ wl,
                       const float* __restrict__ bl, float* __restrict__ h) {
  int p = blockIdx.x * 256 + threadIdx.x;
  if (p >= NOUT) return;
  int x = p / (DD * DD), y = (p / DD) % DD, z = p % DD;
  size_t sp = ((size_t)x * SS + y) * SS + z;
  const float inv = 1.0f / 47.0f;
  float f0 = df[p], f1 = df[NOUT + p];
  float f2 = x * inv, f3 = y * inv, f4 = z * inv;
  for (int o = 0; o < CH; ++o) {
    const float* w = wl + o * 5;
    float acc = bl[o] + w[0] * f0 + w[1] * f1 + w[2] * f2 + w[3] * f3 + w[4] * f4;
    h[(size_t)o * S3P + sp] = acc;
  }
}

// ---------------- conv1x1 GEMM: Y(96 x S3P) = Wpad(96x96) @ X(96 x S3P) + b, optional GELU
// Block tile: 96 rows x 256 cols. 8 waves, each wave owns 2 col-tiles x 6 row-tiles.
// Per K-step: load 2 B-frags + 6 A-frags, then 12 back-to-back WMMAs (one dscnt wait).
__global__ __launch_bounds__(256) void conv_gemm_k(
    const float* __restrict__ W86, const float* __restrict__ b86,
    const float* __restrict__ X, float* __restrict__ Y, int apply_gelu) {
  extern __shared__ float sm[];
  float* Ws = sm;                    // CP*CP
  float* Xs = sm + CP * CP;          // CP*NT_COLS
  float* Bs = Xs + CP * NT_COLS;     // CP
  int tid = threadIdx.x;
  for (int i = tid; i < CP * CP; i += 256) {
    int m = i / CP, k = i % CP;
    Ws[i] = (m < CH && k < CH) ? W86[m * CH + k] : 0.0f;
  }
  if (tid < CP) Bs[tid] = (tid < CH) ? b86[tid] : 0.0f;
  size_t col0 = (size_t)blockIdx.x * NT_COLS;
  for (int i = tid; i < CP * NT_COLS; i += 256) {
    int k = i >> 8, n = i & 255;
    Xs[i] = X[(size_t)k * S3P + col0 + n];
  }
  __syncthreads();
  int lane = tid & 31, wave = tid >> 5;
  int arow = lane & 15, koff = (lane >> 4) << 1;
  int nc0 = wave * 32 + (lane & 15);   // first of two col-tiles per wave
  v8f acc[12];                         // [mt][nt] -> acc[mt*2+nt]
#pragma unroll
  for (int t = 0; t < 12; ++t)
#pragma unroll
    for (int v = 0; v < 8; ++v) acc[t][v] = 0.0f;
  for (int k = 0; k < CP; k += 4) {
    v2f b0, b1;
    b0.x = Xs[(k + koff) * NT_COLS + nc0];
    b0.y = Xs[(k + koff + 1) * NT_COLS + nc0];
    b1.x = Xs[(k + koff) * NT_COLS + nc0 + 16];
    b1.y = Xs[(k + koff + 1) * NT_COLS + nc0 + 16];
    v2f a[6];
#pragma unroll
    for (int mt = 0; mt < 6; ++mt) {
      a[mt].x = Ws[(mt * 16 + arow) * CP + k + koff];
      a[mt].y = Ws[(mt * 16 + arow) * CP + k + koff + 1];
    }
#pragma unroll
    for (int mt = 0; mt < 6; ++mt) {
      acc[mt * 2 + 0] = wmma4(a[mt], b0, acc[mt * 2 + 0]);
      acc[mt * 2 + 1] = wmma4(a[mt], b1, acc[mt * 2 + 1]);
    }
  }
  int rbase = (lane < 16) ? 0 : 8;
#pragma unroll
  for (int mt = 0; mt < 6; ++mt) {
#pragma unroll
    for (int nt = 0; nt < 2; ++nt) {
#pragma unroll
      for (int v = 0; v < 8; ++v) {
        int row = mt * 16 + rbase + v;
        float d = acc[mt * 2 + nt][v] + Bs[row];
        if (apply_gelu) d = gelu_f(d);
        Y[(size_t)row * S3P + col0 + nc0 + nt * 16] = d;
      }
    }
  }
}

// ---------------- truncated separable DFT chain
__global__ void dft_z_k(const float* __restrict__ h, const float* __restrict__ tw,
                        float* __restrict__ out) {
  int idx = blockIdx.x * 256 + threadIdx.x;
  if (idx >= CP * 2916 * 6) return;
  int kz = idx % 6;
  int xy = (idx / 6) % 2916;
  int c = idx / (6 * 2916);
  const float* row = h + (size_t)c * S3P + (size_t)xy * SS;
  float re = 0.f, im = 0.f;
  for (int z = 0; z < SS; ++z) {
    float v = row[z];
    float cs = tw[(z * 12 + kz) * 2], sn = tw[(z * 12 + kz) * 2 + 1];
    re += v * cs;
    im -= v * sn;
  }
  out[2 * idx] = re;
  out[2 * idx + 1] = im;
}

__global__ void dft_y_k(const float* __restrict__ in, const float* __restrict__ tw,
                        float* __restrict__ out) {
  int idx = blockIdx.x * 256 + threadIdx.x;
  if (idx >= CP * 54 * 12 * 6) return;
  int kz = idx % 6;
  int ky = (idx / 6) % 12;
  int x = (idx / 72) % 54;
  int c = idx / (72 * 54);
  const float* base = in + ((size_t)(c * 2916 + x * 54) * 6 + kz) * 2;
  float re = 0.f, im = 0.f;
  for (int y = 0; y < SS; ++y) {
    float br = base[y * 12], bi = base[y * 12 + 1];
    float cs = tw[(y * 12 + ky) * 2], sn = tw[(y * 12 + ky) * 2 + 1];
    re += br * cs + bi * sn;
    im += bi * cs - br * sn;
  }
  out[2 * idx] = re;
  out[2 * idx + 1] = im;
}

__global__ void dft_x_k(const float* __restrict__ in, const float* __restrict__ tw,
                        float* __restrict__ out) {
  int idx = blockIdx.x * 256 + threadIdx.x;
  if (idx >= CP * 864) return;
  int kz = idx % 6;
  int ky = (idx / 6) % 12;
  int kx = (idx / 72) % 12;
  int c = idx / 864;
  const float* base = in + (((size_t)c * 54 * 12 + ky) * 6 + kz) * 2;
  float re = 0.f, im = 0.f;
  for (int x = 0; x < SS; ++x) {
    float br = base[x * 144], bi = base[x * 144 + 1];
    float cs = tw[(x * 12 + kx) * 2], sn = tw[(x * 12 + kx) * 2 + 1];
    re += br * cs + bi * sn;
    im += bi * cs - br * sn;
  }
  out[2 * idx] = re;
  out[2 * idx + 1] = im;
}

// channel mixing per frequency: G[o][f] = sum_i F[i][f] * (wr + i*wi)[blk][i][o][m3]
__global__ void mix_k(const float* __restrict__ F, const float* __restrict__ wr,
                      const float* __restrict__ wi, float* __restrict__ G) {
  int idx = blockIdx.x * 256 + threadIdx.x;
  if (idx >= CP * 864) return;
  int f = idx % 864;
  int o = idx / 864;
  if (o >= CH) {
    G[2 * idx] = 0.f;
    G[2 * idx + 1] = 0.f;
    return;
  }
  int kx = f / 72, ky = (f / 6) % 12, kz = f % 6;
  int blk = ((kx >= 6) ? 2 : 0) + ((ky >= 6) ? 1 : 0);
  int m3 = (kx % 6) * 36 + (ky % 6) * 6 + kz;
  size_t wbase = ((size_t)blk * 7396 + o) * 216 + m3;
  const size_t istr = (size_t)CH * 216;
  float ar = 0.f, ai = 0.f;
  for (int i = 0; i < CH; ++i) {
    float br = F[(size_t)(i * 864 + f) * 2], bi = F[(size_t)(i * 864 + f) * 2 + 1];
    float wrv = wr[wbase + (size_t)i * istr];
    float wiv = wi[wbase + (size_t)i * istr];
    ar += br * wrv - bi * wiv;
    ai += br * wiv + bi * wrv;
  }
  G[2 * idx] = ar;
  G[2 * idx + 1] = ai;
}

__global__ void idft_x_k(const float* __restrict__ in, const float* __restrict__ tw,
                         float* __restrict__ out) {
  int idx = blockIdx.x * 256 + threadIdx.x;
  if (idx >= CP * 54 * 12 * 6) return;
  int kz = idx % 6;
  int ky = (idx / 6) % 12;
  int x = (idx / 72) % 54;
  int c = idx / (72 * 54);
  const float* base = in + ((size_t)c * 864 + ky * 6 + kz) * 2;
  float re = 0.f, im = 0.f;
  for (int kx = 0; kx < 12; ++kx) {
    float gr = base[kx * 144], gi = base[kx * 144 + 1];
    float cs = tw[(x * 12 + kx) * 2], sn = tw[(x * 12 + kx) * 2 + 1];
    re += gr * cs - gi * sn;
    im += gr * sn + gi * cs;
  }
  out[2 * idx] = re;
  out[2 * idx + 1] = im;
}

__global__ void idft_y_k(const float* __restrict__ in, const float* __restrict__ tw,
                         float* __restrict__ out) {
  int idx = blockIdx.x * 256 + threadIdx.x;
  if (idx >= CP * 2916 * 6) return;
  int kz = idx % 6;
  int xy = (idx / 6) % 2916;
  int c = idx / (6 * 2916);
  int y = xy % 54, x = xy / 54;
  const float* base = in + (((size_t)(c * 54 + x) * 12) * 6 + kz) * 2;
  float re = 0.f, im = 0.f;
  for (int ky = 0; ky < 12; ++ky) {
    float br = base[ky * 12], bi = base[ky * 12 + 1];
    float cs = tw[(y * 12 + ky) * 2], sn = tw[(y * 12 + ky) * 2 + 1];
    re += br * cs - bi * sn;
    im += br * sn + bi * cs;
  }
  out[2 * idx] = re;
  out[2 * idx + 1] = im;
}

__global__ void idft_z_k(const float* __restrict__ in, const float* __restrict__ tw,
                         float* __restrict__ out) {
  int idx = blockIdx.x * 256 + threadIdx.x;
  if (idx >= CP * S3) return;
  int z = idx % 54;
  int xy = (idx / 54) % 2916;
  int c = idx / (54 * 2916);
  const float* base = in + (size_t)(c * 2916 + xy) * 12;
  float v = base[0]; // kz = 0 term
#pragma unroll
  for (int kz = 1; kz < 6; ++kz) {
    float re = base[2 * kz], im = base[2 * kz + 1];
    float cs = tw[(z * 12 + kz) * 2], sn = tw[(z * 12 + kz) * 2 + 1];
    v += 2.0f * (re * cs - im * sn);
  }
  out[(size_t)c * S3P + (size_t)xy * SS + z] = v * (1.0f / 157464.0f);
}

// ---------------- group norm (whole-tensor mean/var over c<86, s<S3)
__global__ void gn_partial_k(const float* __restrict__ X, double* __restrict__ part) {
  __shared__ double s1[256];
  __shared__ double s2[256];
  int tid = threadIdx.x;
  const size_t TOT = (size_t)CH * S3;
  double a = 0.0, b = 0.0;
  for (size_t i = (size_t)blockIdx.x * 256 + tid; i < TOT; i += (size_t)864 * 256) {
    int c = (int)(i / S3);
    int s = (int)(i - (size_t)c * S3);
    float v = X[(size_t)c * S3P + s];
    a += v;
    b += (double)v * v;
  }
  s1[tid] = a;
  s2[tid] = b;
  __syncthreads();
  for (int st = 128; st > 0; st >>= 1) {
    if (tid < st) { s1[tid] += s1[tid + st]; s2[tid] += s2[tid + st]; }
    __syncthreads();
  }
  if (tid == 0) {
    part[blockIdx.x * 2] = s1[0];
    part[blockIdx.x * 2 + 1] = s2[0];
  }
}

__global__ void gn_final_k(const double* __restrict__ part, float* __restrict__ stats) {
  __shared__ double s1[256];
  __shared__ double s2[256];
  int tid = threadIdx.x;
  double a = 0.0, b = 0.0;
  for (int i = tid; i < 864; i += 256) { a += part[2 * i]; b += part[2 * i + 1]; }
  s1[tid] = a;
  s2[tid] = b;
  __syncthreads();
  for (int st = 128; st > 0; st >>= 1) {
    if (tid < st) { s1[tid] += s1[tid + st]; s2[tid] += s2[tid + st]; }
    __syncthreads();
  }
  if (tid == 0) {
    double N = (double)CH * (double)S3;
    double mu = s1[0] / N;
    double var = s2[0] / N - mu * mu;
    stats[0] = (float)mu;
    stats[1] = (float)(1.0 / sqrt(var + 1e-5));
  }
}

__global__ void gn_apply_k(const float* xs, const float* skip, const float* __restrict__ stats,
                           const float* __restrict__ gw, const float* __restrict__ gb,
                           float* outp, int dg) {
  size_t idx = (size_t)blockIdx.x * 256 + threadIdx.x;
  int c = (int)(idx / S3P);
  float y = 0.0f;
  if (c < CH) {
    float mu = stats[0], rs = stats[1];
    float hn = (xs[idx] - mu) * rs;
    y = hn * gw[c] + gb[c] + skip[idx];
    if (dg) y = gelu_f(y);
  }
  outp[idx] = y;
}

// ---------------- GNO: one workgroup per query point, WMMA MLP 96->512->256->96
#define XS_ST 100
#define H1_ST 516
#define H2_ST 260
#define H3_ST 96
#define GNO_SMEM_F (32 * XS_ST + 32 * H1_ST + 32 * H2_ST + 32 * H3_ST)

__global__ __launch_bounds__(256) void gno_k(
    const int* __restrict__ nb_idx, const unsigned char* __restrict__ nb_mask,
    const float* __restrict__ xoute, const float* __restrict__ xine,
    const float* __restrict__ k1t, const float* __restrict__ k1b,
    const float* __restrict__ k2t, const float* __restrict__ k2b,
    const float* __restrict__ k3t, const float* __restrict__ k3b,
    const float* __restrict__ h, float* __restrict__ gout) {
  extern __shared__ float sm[];
  float* Xs = sm;
  float* H1 = Xs + 32 * XS_ST;
  float* H2 = H1 + 32 * H1_ST;
  float* H3 = H2 + 32 * H2_ST;
  int p = blockIdx.x;
  int tid = threadIdx.x;
  for (int i = tid; i < 32 * 96; i += 256) {
    int j = i / 96, c = i % 96;
    float v = 0.0f;
    if (j < KNB) {
      if (c < 48) v = xoute[(size_t)nb_idx[p * KNB + j] * 48 + c];
      else        v = xine[(size_t)p * 48 + (c - 48)];
    }
    Xs[j * XS_ST + c] = v;
  }
  __syncthreads();
  int lane = tid & 31, wave = tid >> 5;
  int arow = lane & 15, koff = (lane >> 4) << 1, ncl = lane & 15;
  int rbase = (lane < 16) ? 0 : 8;
  // H1 = gelu(X @ k1^T + b1) : M=32, N=512, K=96; both M-tiles per N-tile share B.
  for (int nt = wave; nt < 32; nt += 8) {
    int n0 = nt * 16 + ncl;
    v8f acc0, acc1;
    float bc = k1b[n0];
#pragma unroll
    for (int v = 0; v < 8; ++v) { acc0[v] = bc; acc1[v] = bc; }
    for (int k = 0; k < 96; k += 4) {
      v2f b, a0, a1;
      b.x = k1t[(k + koff) * 512 + n0];
      b.y = k1t[(k + koff + 1) * 512 + n0];
      a0.x = Xs[arow * XS_ST + k + koff];
      a0.y = Xs[arow * XS_ST + k + koff + 1];
      a1.x = Xs[(16 + arow) * XS_ST + k + koff];
      a1.y = Xs[(16 + arow) * XS_ST + k + koff + 1];
      acc0 = wmma4(a0, b, acc0);
      acc1 = wmma4(a1, b, acc1);
    }
#pragma unroll
    for (int v = 0; v < 8; ++v) {
      H1[(rbase + v) * H1_ST + n0] = gelu_f(acc0[v]);
      H1[(16 + rbase + v) * H1_ST + n0] = gelu_f(acc1[v]);
    }
  }
  __syncthreads();
  // H2 = gelu(H1 @ k2^T + b2) : M=32, N=256, K=512
  for (int nt = wave; nt < 16; nt += 8) {
    int n0 = nt * 16 + ncl;
    v8f acc0, acc1;
    float bc = k2b[n0];
#pragma unroll
    for (int v = 0; v < 8; ++v) { acc0[v] = bc; acc1[v] = bc; }
    for (int k = 0; k < 512; k += 4) {
      v2f b, a0, a1;
      b.x = k2t[(k + koff) * 256 + n0];
      b.y = k2t[(k + koff + 1) * 256 + n0];
      a0.x = H1[arow * H1_ST + k + koff];
      a0.y = H1[arow * H1_ST + k + koff + 1];
      a1.x = H1[(16 + arow) * H1_ST + k + koff];
      a1.y = H1[(16 + arow) * H1_ST + k + koff + 1];
      acc0 = wmma4(a0, b, acc0);
      acc1 = wmma4(a1, b, acc1);
    }
#pragma unroll
    for (int v = 0; v < 8; ++v) {
      H2[(rbase + v) * H2_ST + n0] = gelu_f(acc0[v]);
      H2[(16 + rbase + v) * H2_ST + n0] = gelu_f(acc1[v]);
    }
  }
  __syncthreads();
  // H3 = H2 @ k3^T + b3 : M=32, N=96, K=256
  for (int nt = wave; nt < 6; nt += 8) {
    int n0 = nt * 16 + ncl;
    v8f acc0, acc1;
    float bc = (n0 < CH) ? k3b[n0] : 0.0f;
#pragma unroll
    for (int v = 0; v < 8; ++v) { acc0[v] = bc; acc1[v] = bc; }
    for (int k = 0; k < 256; k += 4) {
      v2f b, a0, a1;
      b.x = k3t[(k + koff) * 96 + n0];
      b.y = k3t[(k + koff + 1) * 96 + n0];
      a0.x = H2[arow * H2_ST + k + koff];
      a0.y = H2[arow * H2_ST + k + koff + 1];
      a1.x = H2[(16 + arow) * H2_ST + k + koff];
      a1.y = H2[(16 + arow) * H2_ST + k + koff + 1];
      acc0 = wmma4(a0, b, acc0);
      acc1 = wmma4(a1, b, acc1);
    }
#pragma unroll
    for (int v = 0; v < 8; ++v) {
      H3[(rbase + v) * H3_ST + n0] = acc0[v];
      H3[(16 + rbase + v) * H3_ST + n0] = acc1[v];
    }
  }
  __syncthreads();
  // masked neighbor reduction against f_y (read directly from padded h buffer)
  if (tid < CP) {
    int c = tid;
    float num = 0.0f;
    int cnt = 0;
    for (int j = 0; j < KNB; ++j) {
      int mk = nb_mask[p * KNB + j];
      if (mk) {
        ++cnt;
        if (c < CH) {
          int g = nb_idx[p * KNB + j];
          int x = g / (DD * DD), y = (g / DD) % DD, z = g % DD;
          size_t sidx = ((size_t)x * SS + y) * SS + z;
          num += H3[j * H3_ST + c] * h[(size_t)c * S3P + sidx];
        }
      }
    }
    float den = (float)((cnt > 0) ? cnt : 1);
    gout[(size_t)p * CP + c] = (c < CH) ? (num / den) : 0.0f;
  }
}

// ---------------- final projection 86 -> 256 -> 1
__global__ __launch_bounds__(256) void proj_k(const float* __restrict__ gno,
                                              const float* __restrict__ p1w,
                                              const float* __restrict__ p1b,
                                              const float* __restrict__ p2w,
                                              const float* __restrict__ p2b,
                                              float* __restrict__ out) {
  __shared__ float v[CH];
  __shared__ float red[256];
  int p = blockIdx.x, tid = threadIdx.x;
  if (tid < CH) v[tid] = gno[(size_t)p * CP + tid];
  __syncthreads();
  float acc = p1b[tid];
  for (int c = 0; c < CH; ++c) acc += p1w[tid * CH + c] * v[c];
  acc = gelu_f(acc);
  red[tid] = acc * p2w[tid];
  __syncthreads();
  for (int s = 128; s > 0; s >>= 1) {
    if (tid < s) red[tid] += red[tid + s];
    __syncthreads();
  }
  if (tid == 0) out[p] = red[0] + p2b[0];
}

extern "C" void kernel_launch(void* const* d_in, const int* in_sizes, int n_in,
                              void* d_out, int out_size, void* d_ws, size_t ws_size,
                              hipStream_t stream) {
  (void)in_sizes; (void)n_in; (void)out_size; (void)ws_size;
  const float* x_in    = (const float*)d_in[0];
  const float* df      = (const float*)d_in[2];
  const int* nb_idx    = (const int*)d_in[3];
  const unsigned char* nb_mask = (const unsigned char*)d_in[4];
  const float* w_lift  = (const float*)d_in[5];
  const float* b_lift  = (const float*)d_in[6];
  const float* spec_wr = (const float*)d_in[7];
  const float* spec_wi = (const float*)d_in[8];
  const float* w_skip  = (const float*)d_in[9];
  const float* b_skip  = (const float*)d_in[10];
  const float* mlp1_w  = (const float*)d_in[11];
  const float* mlp1_b  = (const float*)d_in[12];
  const float* mlp2_w  = (const float*)d_in[13];
  const float* mlp2_b  = (const float*)d_in[14];
  const float* wms     = (const float*)d_in[15];
  const float* bms     = (const float*)d_in[16];
  const float* g1w     = (const float*)d_in[17];
  const float* g1b     = (const float*)d_in[18];
  const float* g2w     = (const float*)d_in[19];
  const float* g2b     = (const float*)d_in[20];
  const float* k1w     = (const float*)d_in[21];
  const float* k1b     = (const float*)d_in[22];
  const float* k2w     = (const float*)d_in[23];
  const float* k2b     = (const float*)d_in[24];
  const float* k3w     = (const float*)d_in[25];
  const float* k3b     = (const float*)d_in[26];
  const float* p1w     = (const float*)d_in[27];
  const float* p1b     = (const float*)d_in[28];
  const float* p2w     = (const float*)d_in[29];
  const float* p2b     = (const float*)d_in[30];
  float* out = (float*)d_out;

  float* ws = (float*)d_ws;
  size_t o = 0;
  float* bufH = ws + o; o += (size_t)CP * S3P;
  float* bufS = ws + o; o += (size_t)CP * S3P;
  float* bufT = ws + o; o += (size_t)CP * S3P;
  float* cA   = ws + o; o += (size_t)CP * 2916 * 6 * 2;
  float* cB   = ws + o; o += (size_t)CP * 54 * 12 * 6 * 2;
  float* cC   = ws + o; o += (size_t)CP * 864 * 2;
  float* cD   = ws + o; o += (size_t)CP * 864 * 2;
  float* tw   = ws + o; o += 54 * 12 * 2;
  float* xoe  = ws + o; o += (size_t)NOUT * 48;
  float* xie  = ws + o; o += (size_t)NIN * 48;
  float* k1t  = ws + o; o += 96 * 512;
  float* k2t  = ws + o; o += 512 * 256;
  float* k3t  = ws + o; o += 256 * 96;
  float* gno  = ws + o; o += (size_t)NIN * CP;
  double* part = (double*)(ws + o); o += 864 * 4;
  float* stats = ws + o; o += 8;

  hipMemsetAsync(bufH, 0, (size_t)CP * S3P * sizeof(float), stream);
  twiddle_init_k<<<3, 256, 0, stream>>>(tw);
  embed_out_k<<<NOUT / 256, 256, 0, stream>>>(xoe);
  embed_in_k<<<NIN / 256, 256, 0, stream>>>(x_in, xie);
  transpose_pad_k<<<(96 * 512 + 255) / 256, 256, 0, stream>>>(k1w, k1t, 512, 96, 512);
  transpose_pad_k<<<(512 * 256 + 255) / 256, 256, 0, stream>>>(k2w, k2t, 256, 512, 256);
  transpose_pad_k<<<(256 * 96 + 255) / 256, 256, 0, stream>>>(k3w, k3t, 86, 256, 96);
  lift_k<<<NOUT / 256, 256, 0, stream>>>(df, w_lift, b_lift, bufH);

  const size_t gemm_lds = (size_t)(CP * CP + CP * NT_COLS + CP) * sizeof(float);
  for (int l = 0; l < 4; ++l) {
    const size_t wo = (size_t)l * CH * CH;
    const size_t bo = (size_t)l * CH;
    const size_t so = (size_t)l * 4 * CH * CH * 216;
    const int dg = (l < 3) ? 1 : 0;
    conv_gemm_k<<<NBLK_GEMM, 256, gemm_lds, stream>>>(w_skip + wo, b_skip + bo, bufH, bufS, 0);
    dft_z_k<<<CP * 2916 * 6 / 256, 256, 0, stream>>>(bufH, tw, cA);
    dft_y_k<<<CP * 54 * 12 * 6 / 256, 256, 0, stream>>>(cA, tw, cB);
    dft_x_k<<<CP * 864 / 256, 256, 0, stream>>>(cB, tw, cC);
    mix_k<<<CP * 864 / 256, 256, 0, stream>>>(cC, spec_wr + so, spec_wi + so, cD);
    idft_x_k<<<CP * 54 * 12 * 6 / 256, 256, 0, stream>>>(cD, tw, cB);
    idft_y_k<<<CP * 2916 * 6 / 256, 256, 0, stream>>>(cB, tw, cA);
    idft_z_k<<<CP * S3 / 256, 256, 0, stream>>>(cA, tw, bufT);
    gn_partial_k<<<864, 256, 0, stream>>>(bufT, part);
    gn_final_k<<<1, 256, 0, stream>>>(part, stats);
    gn_apply_k<<<(size_t)CP * S3P / 256, 256, 0, stream>>>(bufT, bufS, stats, g1w + bo, g1b + bo, bufH, dg);
    conv_gemm_k<<<NBLK_GEMM, 256, gemm_lds, stream>>>(wms + wo, bms + bo, bufH, bufS, 0);
    conv_gemm_k<<<NBLK_GEMM, 256, gemm_lds, stream>>>(mlp1_w + wo, mlp1_b + bo, bufH, bufT, 1);
    conv_gemm_k<<<NBLK_GEMM, 256, gemm_lds, stream>>>(mlp2_w + wo, mlp2_b + bo, bufT, bufH, 0);
    gn_partial_k<<<864, 256, 0, stream>>>(bufH, part);
    gn_final_k<<<1, 256, 0, stream>>>(part, stats);
    gn_apply_k<<<(size_t)CP * S3P / 256, 256, 0, stream>>>(bufH, bufS, stats, g2w + bo, g2b + bo, bufH, dg);
  }
  const size_t gno_lds = (size_t)GNO_SMEM_F * sizeof(float);
  gno_k<<<NIN, 256, gno_lds, stream>>>(nb_idx, nb_mask, xoe, xie, k1t, k1b, k2t, k2b,
                                       k3t, k3b, bufH, gno);
  proj_k<<<NIN, 256, 0, stream>>>(gno, p1w, p1b, p2w, p2b, out);
}